// WaveCore_74818330296780
// MI455X (gfx1250) — compile-verified
//
#include <hip/hip_runtime.h>

// CDNA5 (gfx1250) wave32 WMMA types
typedef float v2f __attribute__((ext_vector_type(2)));
typedef float v8f __attribute__((ext_vector_type(8)));

__device__ __forceinline__ v8f wmma4(v2f a, v2f b, v8f c) {
    // D = A(16x4,f32) x B(4x16,f32) + C(16x16,f32)
    return __builtin_amdgcn_wmma_f32_16x16x4_f32(
        /*neg_a=*/false, a, /*neg_b=*/false, b,
        /*c_mod=*/(short)0, c, /*reuse_a=*/false, /*reuse_b=*/false);
}

#define ROWS      16
#define XSTRIDE   1080   // floats per row in padded-x slab; x at [8,1032)
#define CSTRIDE   520    // floats per row in coeff slabs; j index stored at 4+j
#define LC        516    // coefficients per filter branch

extern "C" __global__ __launch_bounds__(256)
void wavecore_kernel(const float* __restrict__ x,
                     const float* __restrict__ dec_lo,
                     const float* __restrict__ dec_hi,
                     const float* __restrict__ rec_lo,
                     const float* __restrict__ rec_hi,
                     const float* __restrict__ sp_c,
                     const int*   __restrict__ sp_i,
                     float* __restrict__ out)
{
    extern __shared__ float smem[];
    float* xb  = smem;                       // ROWS * XSTRIDE
    float* loe = xb  + ROWS * XSTRIDE;       // ROWS * CSTRIDE  (lo, guard j=-4..-1 at [0,4))
    float* hie = loe + ROWS * CSTRIDE;       // ROWS * CSTRIDE
    float* flt = hie + ROWS * CSTRIDE;       // 32: [0..7]=dec_lo flipped, [8..15]=dec_hi flipped,
                                             //     [16..23]=rec_lo, [24..31]=rec_hi

    const int tid  = threadIdx.x;
    const int lane = tid & 31;
    const int wv   = tid >> 5;        // 0..7 (8 wave32 per block)
    const int half = lane >> 4;       // lane[4]
    const int l    = lane & 15;       // lane%16
    const size_t row0 = (size_t)blockIdx.x * ROWS;

    // ---- stage 1: async DMA of x slab into LDS (ASYNCcnt path) ----------
    // Generic AS(3) pointers carry the LDS byte offset in the low 32 bits.
    {
        const unsigned ldsbase = (unsigned)(uintptr_t)xb;
        const float* xbase = x + row0 * 1024;   // wave-uniform -> SGPR pair
        for (int v = tid; v < ROWS * 256; v += 256) {
            int r = v >> 8, c = v & 255;
            unsigned dst  = ldsbase + (unsigned)((r * XSTRIDE + 8 + 4 * c) * 4);
            unsigned voff = (unsigned)v << 4;   // (r*1024 + 4*c)*4 bytes
            asm volatile("global_load_async_to_lds_b128 %0, %1, %2"
                         :: "v"(dst), "v"(voff), "s"(xbase) : "memory");
        }
    }
    if (tid < 8) {
        flt[tid]      = dec_lo[7 - tid];   // flipped (conv with w[::-1])
        flt[8 + tid]  = dec_hi[7 - tid];
        flt[16 + tid] = rec_lo[tid];
        flt[24 + tid] = rec_hi[tid];
    }
    // zero tail [1032, XSTRIDE); reflect fill overwrites [1032,1039) later
    for (int v = tid; v < ROWS * (XSTRIDE - 1032); v += 256) {
        int r = v / (XSTRIDE - 1032);
        int c = 1032 + v % (XSTRIDE - 1032);
        xb[r * XSTRIDE + c] = 0.f;
    }
    // zero j<0 guard cells of coefficient slabs
    if (tid < ROWS * 4) {
        int r = tid >> 2, c = tid & 3;
        loe[r * CSTRIDE + c] = 0.f;
        hie[r * CSTRIDE + c] = 0.f;
    }
    // all of this wave's async LDS writes must land before the barrier
    asm volatile("s_wait_asynccnt 0x0" ::: "memory");
    __syncthreads();

    // ---- stage 2: reflect padding of x (xp[u] = xb[u+1]) -----------------
    if (tid < ROWS * 16) {
        int r = tid >> 4, k = tid & 15;
        if (k < 7) {
            xb[r * XSTRIDE + 1 + k] = xb[r * XSTRIDE + 15 - k];        // left:  x[7-u]
        } else if (k < 14) {
            int k2 = k - 7;                                            // right: x[1022-k2]
            xb[r * XSTRIDE + 1032 + k2] = xb[r * XSTRIDE + 1030 - k2];
        }
    }

    // ---- per-lane constant B fragments (filters only; tile-independent) --
    // Forward: W[t][n] = wf[t-2n], t = 4*tc + 2*half (+0/+1), n = l
    v2f bLo[10], bHi[10];
#pragma unroll
    for (int tc = 0; tc < 10; ++tc) {
        int t  = 4 * tc + 2 * half;
        int d0 = t - 2 * l, d1 = d0 + 1;
        bLo[tc].x = (d0 >= 0 && d0 < 8) ? flt[d0]     : 0.f;
        bLo[tc].y = (d1 >= 0 && d1 < 8) ? flt[d1]     : 0.f;
        bHi[tc].x = (d0 >= 0 && d0 < 8) ? flt[8 + d0] : 0.f;
        bHi[tc].y = (d1 >= 0 && d1 < 8) ? flt[8 + d1] : 0.f;
    }
    // Inverse: WL[r][n] = rec_lo[2r-n-1], r = 4*tc + 2*half (+0/+1)
    v2f bRL[4], bRH[4];
#pragma unroll
    for (int tc = 0; tc < 4; ++tc) {
        int r0 = 4 * tc + 2 * half;
        int d0 = 2 * r0 - l - 1, d1 = d0 + 2;
        bRL[tc].x = (d0 >= 0 && d0 < 8) ? flt[16 + d0] : 0.f;
        bRL[tc].y = (d1 >= 0 && d1 < 8) ? flt[16 + d1] : 0.f;
        bRH[tc].x = (d0 >= 0 && d0 < 8) ? flt[24 + d0] : 0.f;
        bRH[tc].y = (d1 >= 0 && d1 < 8) ? flt[24 + d1] : 0.f;
    }
    __syncthreads();

    // ---- stage 3: forward DWT via WMMA (66 tiles of 16 outputs) ----------
    for (int tile = wv; tile < 66; tile += 8) {
        const int  isHi = (tile >= 33);
        const int  j0   = (isHi ? tile - 33 : tile) * 16;
        const float* xrow = xb + l * XSTRIDE + 2 * j0 + 2 * half + 1; // xp index +1
        v8f acc = {};
        if (!isHi) {
#pragma unroll
            for (int tc = 0; tc < 10; ++tc) {
                v2f a; a.x = xrow[4 * tc]; a.y = xrow[4 * tc + 1];
                acc = wmma4(a, bLo[tc], acc);
            }
        } else {
#pragma unroll
            for (int tc = 0; tc < 10; ++tc) {
                v2f a; a.x = xrow[4 * tc]; a.y = xrow[4 * tc + 1];
                acc = wmma4(a, bHi[tc], acc);
            }
        }
        float* tgt = isHi ? hie : loe;
        int j = j0 + l;                 // output coefficient column
        if (j < LC) {
#pragma unroll
            for (int g = 0; g < 8; ++g)
                tgt[(g + 8 * half) * CSTRIDE + 4 + j] = acc[g];
        }
    }
    __syncthreads();

    // ---- stage 4: sparse coefficient adds (same indices for every row) ---
    if (tid < ROWS * 10) {
        int r = tid / 10, s = tid - r * 10;
        int idx = sp_i[s];
        float c = sp_c[s];
        if (idx < LC) loe[r * CSTRIDE + 4 + idx] += c;
        else          hie[r * CSTRIDE + 4 + (idx - LC)] += c;
    }
    __syncthreads();

    // ---- stage 5: inverse DWT via WMMA + fused x-add (64 tiles) ----------
    for (int tile = wv; tile < 64; tile += 8) {
        const int i0 = tile * 16;
        // lo_ext slab index = (jb + r) + 4 = 8*tile + r   (jb = 8*tile - 4)
        const float* lrow = loe + l * CSTRIDE + 8 * tile + 2 * half;
        const float* hrow = hie + l * CSTRIDE + 8 * tile + 2 * half;
        v8f acc = {};
#pragma unroll
        for (int tc = 0; tc < 4; ++tc) {
            v2f a; a.x = lrow[4 * tc]; a.y = lrow[4 * tc + 1];
            acc = wmma4(a, bRL[tc], acc);
            v2f b; b.x = hrow[4 * tc]; b.y = hrow[4 * tc + 1];
            acc = wmma4(b, bRH[tc], acc);
        }
        // left reflect-boundary fixup: i = l in [0,6], reflected taps q<0
        if (tile == 0 && l <= 6) {
            for (int k = 0; k + l <= 6; ++k) {
                int m = 7 - l - k;
                if ((m & 1) == 0) {
                    int jj = m >> 1;
                    float cl = flt[16 + k], ch = flt[24 + k];
#pragma unroll
                    for (int g = 0; g < 8; ++g) {
                        int rr = g + 8 * half;
                        acc[g] += loe[rr * CSTRIDE + 4 + jj] * cl
                                + hie[rr * CSTRIDE + 4 + jj] * ch;
                    }
                }
            }
        }
        // out = x + rec
#pragma unroll
        for (int g = 0; g < 8; ++g) {
            int rr = g + 8 * half;
            float xv = xb[rr * XSTRIDE + 8 + i0 + l];
            out[(row0 + rr) * 1024 + i0 + l] = xv + acc[g];
        }
    }
}

extern "C" void kernel_launch(void* const* d_in, const int* in_sizes, int n_in,
                              void* d_out, int out_size, void* d_ws, size_t ws_size,
                              hipStream_t stream) {
    const float* x      = (const float*)d_in[0];
    const float* dec_lo = (const float*)d_in[1];
    const float* dec_hi = (const float*)d_in[2];
    const float* rec_lo = (const float*)d_in[3];
    const float* rec_hi = (const float*)d_in[4];
    const float* sp_c   = (const float*)d_in[5];
    const int*   sp_i   = (const int*)d_in[6];
    float* out = (float*)d_out;

    int B = (n_in > 0 && in_sizes) ? (in_sizes[0] / 1024) : 32768;
    const size_t shbytes =
        (size_t)(ROWS * XSTRIDE + 2 * ROWS * CSTRIDE + 32) * sizeof(float); // ~133 KB

    hipLaunchKernelGGL(wavecore_kernel, dim3(B / ROWS), dim3(256), shbytes, stream,
                       x, dec_lo, dec_hi, rec_lo, rec_hi, sp_c, sp_i, out);
}